// MambaBlock_87952340288079
// MI455X (gfx1250) — compile-verified
//
#include <hip/hip_runtime.h>
#include <math.h>

#define DMODEL 1024
#define DINNER 2048
#define DSTATE 16
#define DTRANK 64
#define LSEQ   2048
#define CHUNK  128
#define NCHUNK (LSEQ / CHUNK)   // 16

typedef __attribute__((ext_vector_type(2))) float v2f;
typedef __attribute__((ext_vector_type(8))) float v8f;

// ---------------------------------------------------------------------------
// fp32 WMMA GEMM (big tile): C[M,N] = A[M,K](stride LDA) * W[N,K]^T
// Compile-time N/K/LDA so strides fold into instruction immediate offsets.
// Each wave computes a 32(M) x 64(N) tile: 2 A-frags x 4 B-frags -> 8 WMMAs
// per K-step of 4. Requires M%32==0, N%64==0.
// ---------------------------------------------------------------------------
template <int N, int K, int LDA>
__global__ void wmma_gemm_f32_32x64_t(const float* __restrict__ A,
                                      const float* __restrict__ W,
                                      float* __restrict__ C, int M)
{
    const int wavesPerBlock = blockDim.x >> 5;
    const int wave = blockIdx.x * wavesPerBlock + (threadIdx.x >> 5);
    const int lane = threadIdx.x & 31;
    const int ntiles = N >> 6;                 // 64-wide N tiles
    const int mt = wave / ntiles;
    const int nt = wave - mt * ntiles;
    if (mt * 32 >= M) return;                  // uniform per-wave branch

    const int half = lane >> 4;
    const int mr   = lane & 15;

    const float* A0 = A + (size_t)(mt * 32 + mr) * LDA + (half << 1);
    const float* A1 = A0 + (size_t)16 * LDA;
    const float* B0 = W + (size_t)(nt * 64 + mr) * K + (half << 1);
    const float* B1 = B0 + (size_t)16 * K;
    const float* B2 = B0 + (size_t)32 * K;
    const float* B3 = B0 + (size_t)48 * K;

    const v8f vzero = {0.f, 0.f, 0.f, 0.f, 0.f, 0.f, 0.f, 0.f};
    v8f acc[2][4];
#pragma unroll
    for (int i = 0; i < 2; ++i)
#pragma unroll
        for (int j = 0; j < 4; ++j) acc[i][j] = vzero;

#pragma unroll 4
    for (int k = 0; k < K; k += 4) {
        v2f a0 = *(const v2f*)(A0 + k);
        v2f a1 = *(const v2f*)(A1 + k);
        v2f b0 = *(const v2f*)(B0 + k);
        v2f b1 = *(const v2f*)(B1 + k);
        v2f b2 = *(const v2f*)(B2 + k);
        v2f b3 = *(const v2f*)(B3 + k);
        acc[0][0] = __builtin_amdgcn_wmma_f32_16x16x4_f32(false, a0, false, b0, (short)0, acc[0][0], false, false);
        acc[0][1] = __builtin_amdgcn_wmma_f32_16x16x4_f32(false, a0, false, b1, (short)0, acc[0][1], false, false);
        acc[0][2] = __builtin_amdgcn_wmma_f32_16x16x4_f32(false, a0, false, b2, (short)0, acc[0][2], false, false);
        acc[0][3] = __builtin_amdgcn_wmma_f32_16x16x4_f32(false, a0, false, b3, (short)0, acc[0][3], false, false);
        acc[1][0] = __builtin_amdgcn_wmma_f32_16x16x4_f32(false, a1, false, b0, (short)0, acc[1][0], false, false);
        acc[1][1] = __builtin_amdgcn_wmma_f32_16x16x4_f32(false, a1, false, b1, (short)0, acc[1][1], false, false);
        acc[1][2] = __builtin_amdgcn_wmma_f32_16x16x4_f32(false, a1, false, b2, (short)0, acc[1][2], false, false);
        acc[1][3] = __builtin_amdgcn_wmma_f32_16x16x4_f32(false, a1, false, b3, (short)0, acc[1][3], false, false);
    }

#pragma unroll
    for (int i = 0; i < 2; ++i) {
#pragma unroll
        for (int j = 0; j < 4; ++j) {
            float* Crow = C + (size_t)(mt * 32 + i * 16 + half * 8) * N + nt * 64 + j * 16 + mr;
#pragma unroll
            for (int r = 0; r < 8; ++r)
                Crow[(size_t)r * N] = acc[i][j][r];
        }
    }
}

// ---------------------------------------------------------------------------
// fp32 WMMA GEMM (small tile, 16x32 per wave) — used for N=96 (x-proj).
// ---------------------------------------------------------------------------
template <int N, int K, int LDA>
__global__ void wmma_gemm_f32_16x32_t(const float* __restrict__ A,
                                      const float* __restrict__ W,
                                      float* __restrict__ C, int M)
{
    const int wavesPerBlock = blockDim.x >> 5;
    const int wave = blockIdx.x * wavesPerBlock + (threadIdx.x >> 5);
    const int lane = threadIdx.x & 31;
    const int ntiles = N >> 5;
    const int mt = wave / ntiles;
    const int nt = wave - mt * ntiles;
    if (mt * 16 >= M) return;

    const int half = lane >> 4;
    const int mr   = lane & 15;

    const float* Arow = A + (size_t)(mt * 16 + mr) * LDA + (half << 1);
    const float* Wr0  = W + (size_t)(nt * 32 + mr) * K + (half << 1);
    const float* Wr1  = Wr0 + (size_t)16 * K;

    v8f acc0 = {0.f, 0.f, 0.f, 0.f, 0.f, 0.f, 0.f, 0.f};
    v8f acc1 = {0.f, 0.f, 0.f, 0.f, 0.f, 0.f, 0.f, 0.f};

#pragma unroll 4
    for (int k = 0; k < K; k += 4) {
        v2f a  = *(const v2f*)(Arow + k);
        v2f b0 = *(const v2f*)(Wr0 + k);
        v2f b1 = *(const v2f*)(Wr1 + k);
        acc0 = __builtin_amdgcn_wmma_f32_16x16x4_f32(false, a, false, b0, (short)0, acc0, false, false);
        acc1 = __builtin_amdgcn_wmma_f32_16x16x4_f32(false, a, false, b1, (short)0, acc1, false, false);
    }

    float* Crow = C + (size_t)(mt * 16 + half * 8) * N + nt * 32 + mr;
#pragma unroll
    for (int r = 0; r < 8; ++r) {
        Crow[(size_t)r * N]      = acc0[r];
        Crow[(size_t)r * N + 16] = acc1[r];
    }
}

// ---------------------------------------------------------------------------
// Depthwise causal conv1d (k=4, left-pad 3) + bias + SiLU.
// ---------------------------------------------------------------------------
__global__ void conv_silu_k(const float* __restrict__ xar,
                            const float* __restrict__ cw,
                            const float* __restrict__ cb,
                            float* __restrict__ xs)
{
    int idx = blockIdx.x * blockDim.x + threadIdx.x;
    if (idx >= LSEQ * DINNER) return;
    int t = idx / DINNER;
    int d = idx - t * DINNER;
    float acc = cb[d];
    const float* w = cw + d * 4;
#pragma unroll
    for (int j = 0; j < 4; ++j) {
        int tt = t - 3 + j;
        if (tt >= 0)
            acc = fmaf(xar[(size_t)tt * (2 * DINNER) + d], w[j], acc);
    }
    float sig = 1.0f / (1.0f + __expf(-acc));
    xs[idx] = acc * sig;
}

// ---------------------------------------------------------------------------
// In-place delta = softplus(dpre + b_dt)
// ---------------------------------------------------------------------------
__global__ void softplus_k(float* __restrict__ dp, const float* __restrict__ b)
{
    int idx = blockIdx.x * blockDim.x + threadIdx.x;
    if (idx >= LSEQ * DINNER) return;
    int d = idx & (DINNER - 1);
    float x = dp[idx] + b[d];
    dp[idx] = fmaxf(x, 0.0f) + log1pf(__expf(-fabsf(x)));
}

// ---------------------------------------------------------------------------
// Chunked selective scan, phase 1: for each (d, chunk), run the recurrence
// from h=0 over CHUNK steps; record final state Hc and decay product P.
// ---------------------------------------------------------------------------
__global__ void scan_phase1_k(const float* __restrict__ xs,
                              const float* __restrict__ delta,
                              const float* __restrict__ xdbl,
                              const float* __restrict__ A_log,
                              float* __restrict__ Pbuf,
                              float* __restrict__ Hc)
{
    int idx = blockIdx.x * blockDim.x + threadIdx.x;
    if (idx >= DINNER * NCHUNK) return;
    int d = idx & (DINNER - 1);       // consecutive lanes -> consecutive d (coalesced)
    int c = idx >> 11;                // chunk id

    float Ac[DSTATE];
#pragma unroll
    for (int n = 0; n < DSTATE; ++n)
        Ac[n] = -__expf(A_log[d * DSTATE + n]);

    float h[DSTATE], P[DSTATE];
#pragma unroll
    for (int n = 0; n < DSTATE; ++n) { h[n] = 0.0f; P[n] = 1.0f; }

    const int t0 = c * CHUNK;
    for (int t = t0; t < t0 + CHUNK; ++t) {
        float dt = delta[(size_t)t * DINNER + d];
        float u  = xs[(size_t)t * DINNER + d];
        float du = dt * u;
        const float4* Bv = (const float4*)(xdbl + (size_t)t * 96 + DTRANK);
#pragma unroll
        for (int q = 0; q < 4; ++q) {
            float4 B4 = Bv[q];
            float dA;
            dA = __expf(dt * Ac[4*q+0]); P[4*q+0] *= dA; h[4*q+0] = fmaf(dA, h[4*q+0], du * B4.x);
            dA = __expf(dt * Ac[4*q+1]); P[4*q+1] *= dA; h[4*q+1] = fmaf(dA, h[4*q+1], du * B4.y);
            dA = __expf(dt * Ac[4*q+2]); P[4*q+2] *= dA; h[4*q+2] = fmaf(dA, h[4*q+2], du * B4.z);
            dA = __expf(dt * Ac[4*q+3]); P[4*q+3] *= dA; h[4*q+3] = fmaf(dA, h[4*q+3], du * B4.w);
        }
    }

    size_t base = ((size_t)d * NCHUNK + c) * DSTATE;
#pragma unroll
    for (int n = 0; n < DSTATE; ++n) {
        Pbuf[base + n] = P[n];
        Hc[base + n]   = h[n];
    }
}

// ---------------------------------------------------------------------------
// Phase 2: serial combine across chunks per channel (tiny).
// ---------------------------------------------------------------------------
__global__ void scan_combine_k(const float* __restrict__ Pbuf,
                               const float* __restrict__ Hc,
                               float* __restrict__ Hin)
{
    int d = blockIdx.x * blockDim.x + threadIdx.x;
    if (d >= DINNER) return;
    float hr[DSTATE];
#pragma unroll
    for (int n = 0; n < DSTATE; ++n) hr[n] = 0.0f;
    for (int c = 0; c < NCHUNK; ++c) {
        size_t base = ((size_t)d * NCHUNK + c) * DSTATE;
#pragma unroll
        for (int n = 0; n < DSTATE; ++n) {
            Hin[base + n] = hr[n];
            hr[n] = fmaf(Pbuf[base + n], hr[n], Hc[base + n]);
        }
    }
}

// ---------------------------------------------------------------------------
// Phase 3: re-walk each chunk from its exact incoming state, emitting y.
// ---------------------------------------------------------------------------
__global__ void scan_phase3_k(const float* __restrict__ xs,
                              const float* __restrict__ delta,
                              const float* __restrict__ xdbl,
                              const float* __restrict__ A_log,
                              const float* __restrict__ Dv,
                              const float* __restrict__ Hin,
                              float* __restrict__ y)
{
    int idx = blockIdx.x * blockDim.x + threadIdx.x;
    if (idx >= DINNER * NCHUNK) return;
    int d = idx & (DINNER - 1);
    int c = idx >> 11;

    float Ac[DSTATE];
#pragma unroll
    for (int n = 0; n < DSTATE; ++n)
        Ac[n] = -__expf(A_log[d * DSTATE + n]);
    const float Dd = Dv[d];

    float h[DSTATE];
    size_t base = ((size_t)d * NCHUNK + c) * DSTATE;
#pragma unroll
    for (int n = 0; n < DSTATE; ++n) h[n] = Hin[base + n];

    const int t0 = c * CHUNK;
    for (int t = t0; t < t0 + CHUNK; ++t) {
        float dt = delta[(size_t)t * DINNER + d];
        float u  = xs[(size_t)t * DINNER + d];
        float du = dt * u;
        const float4* Bv = (const float4*)(xdbl + (size_t)t * 96 + DTRANK);
        const float4* Cv = (const float4*)(xdbl + (size_t)t * 96 + DTRANK + DSTATE);
        float yacc = 0.0f;
#pragma unroll
        for (int q = 0; q < 4; ++q) {
            float4 B4 = Bv[q];
            float4 C4 = Cv[q];
            float dA;
            dA = __expf(dt * Ac[4*q+0]); h[4*q+0] = fmaf(dA, h[4*q+0], du * B4.x); yacc = fmaf(h[4*q+0], C4.x, yacc);
            dA = __expf(dt * Ac[4*q+1]); h[4*q+1] = fmaf(dA, h[4*q+1], du * B4.y); yacc = fmaf(h[4*q+1], C4.y, yacc);
            dA = __expf(dt * Ac[4*q+2]); h[4*q+2] = fmaf(dA, h[4*q+2], du * B4.z); yacc = fmaf(h[4*q+2], C4.z, yacc);
            dA = __expf(dt * Ac[4*q+3]); h[4*q+3] = fmaf(dA, h[4*q+3], du * B4.w); yacc = fmaf(h[4*q+3], C4.w, yacc);
        }
        y[(size_t)t * DINNER + d] = fmaf(u, Dd, yacc);
    }
}

// ---------------------------------------------------------------------------
// yg = y * silu(res), res = xar[:, DINNER:2*DINNER]
// ---------------------------------------------------------------------------
__global__ void gate_k(const float* __restrict__ y,
                       const float* __restrict__ xar,
                       float* __restrict__ yg)
{
    int idx = blockIdx.x * blockDim.x + threadIdx.x;
    if (idx >= LSEQ * DINNER) return;
    int t = idx / DINNER;
    int d = idx - t * DINNER;
    float r = xar[(size_t)t * (2 * DINNER) + DINNER + d];
    float sig = 1.0f / (1.0f + __expf(-r));
    yg[idx] = y[idx] * (r * sig);
}

// ---------------------------------------------------------------------------
template <int N, int K, int LDA>
static inline void launch_gemm_big(const float* A, const float* W, float* C,
                                   int M, hipStream_t s)
{
    int waves   = (M / 32) * (N / 64);
    int threads = waves * 32;
    int block   = 256;
    int grid    = (threads + block - 1) / block;
    hipLaunchKernelGGL((wmma_gemm_f32_32x64_t<N, K, LDA>), dim3(grid), dim3(block), 0, s,
                       A, W, C, M);
}

template <int N, int K, int LDA>
static inline void launch_gemm_small(const float* A, const float* W, float* C,
                                     int M, hipStream_t s)
{
    int waves   = (M / 16) * (N / 32);
    int threads = waves * 32;
    int block   = 256;
    int grid    = (threads + block - 1) / block;
    hipLaunchKernelGGL((wmma_gemm_f32_16x32_t<N, K, LDA>), dim3(grid), dim3(block), 0, s,
                       A, W, C, M);
}

extern "C" void kernel_launch(void* const* d_in, const int* in_sizes, int n_in,
                              void* d_out, int out_size, void* d_ws, size_t ws_size,
                              hipStream_t stream)
{
    const float* x      = (const float*)d_in[0];  // [1,2048,1024]
    const float* W_in   = (const float*)d_in[1];  // [4096,1024]
    const float* conv_w = (const float*)d_in[2];  // [2048,1,4]
    const float* conv_b = (const float*)d_in[3];  // [2048]
    const float* W_x    = (const float*)d_in[4];  // [96,2048]
    const float* W_dt   = (const float*)d_in[5];  // [2048,64]
    const float* b_dt   = (const float*)d_in[6];  // [2048]
    const float* A_log  = (const float*)d_in[7];  // [2048,16]
    const float* Dp     = (const float*)d_in[8];  // [2048]
    const float* W_out  = (const float*)d_in[9];  // [1024,2048]
    float* out = (float*)d_out;                    // [2048,1024]

    float* ws   = (float*)d_ws;
    float* xar  = ws;                                   // [2048,4096]
    float* xs   = xar  + (size_t)LSEQ * 2 * DINNER;     // [2048,2048]
    float* xdbl = xs   + (size_t)LSEQ * DINNER;         // [2048,96]
    float* dpre = xdbl + (size_t)LSEQ * 96;             // [2048,2048] -> delta (in place)
    float* yb   = dpre + (size_t)LSEQ * DINNER;         // [2048,2048]
    float* Pbuf = yb   + (size_t)LSEQ * DINNER;         // [2048,16,16]
    float* Hc   = Pbuf + (size_t)DINNER * NCHUNK * DSTATE;
    float* Hin  = Hc   + (size_t)DINNER * NCHUNK * DSTATE;
    float* yg   = dpre;                                 // reuse after scan consumes delta

    const int ew_n = LSEQ * DINNER;
    const int ew_block = 256;
    const int ew_grid = (ew_n + ew_block - 1) / ew_block;
    const int sc_n = DINNER * NCHUNK;
    const int sc_grid = (sc_n + ew_block - 1) / ew_block;

    // 1) x_and_res = x @ W_in^T            [2048,4096]
    launch_gemm_big<2 * DINNER, DMODEL, DMODEL>(x, W_in, xar, LSEQ, stream);
    // 2) xs = silu(causal depthwise conv(x_and_res[:, :2048]) + b)
    hipLaunchKernelGGL(conv_silu_k, dim3(ew_grid), dim3(ew_block), 0, stream,
                       xar, conv_w, conv_b, xs);
    // 3) x_dbl = xs @ W_x^T                [2048,96]
    launch_gemm_small<DTRANK + 2 * DSTATE, DINNER, DINNER>(xs, W_x, xdbl, LSEQ, stream);
    // 4) dpre = x_dbl[:, :64] @ W_dt^T     [2048,2048]
    launch_gemm_big<DINNER, DTRANK, 96>(xdbl, W_dt, dpre, LSEQ, stream);
    // 5) delta = softplus(dpre + b_dt)     (in place)
    hipLaunchKernelGGL(softplus_k, dim3(ew_grid), dim3(ew_block), 0, stream,
                       dpre, b_dt);
    // 6) chunked selective scan -> yb
    hipLaunchKernelGGL(scan_phase1_k, dim3(sc_grid), dim3(ew_block), 0, stream,
                       xs, dpre, xdbl, A_log, Pbuf, Hc);
    hipLaunchKernelGGL(scan_combine_k, dim3(DINNER / ew_block), dim3(ew_block), 0, stream,
                       Pbuf, Hc, Hin);
    hipLaunchKernelGGL(scan_phase3_k, dim3(sc_grid), dim3(ew_block), 0, stream,
                       xs, dpre, xdbl, A_log, Dp, Hin, yb);
    // 7) yg = yb * silu(res)
    hipLaunchKernelGGL(gate_k, dim3(ew_grid), dim3(ew_block), 0, stream,
                       yb, xar, yg);
    // 8) out = yg @ W_out^T                [2048,1024]
    launch_gemm_big<DMODEL, DINNER, DINNER>(yg, W_out, out, LSEQ, stream);
}